// GCN_89859305766958
// MI455X (gfx1250) — compile-verified
//
#include <hip/hip_runtime.h>

typedef __attribute__((ext_vector_type(2))) float v2f;
typedef __attribute__((ext_vector_type(8))) float v8f;

#define IN_FEA 256
#define OU_FEA 64

// ---------------------------------------------------------------------------
// Kernel 0: zero the output (harness poisons d_out, scatter uses atomics)
// ---------------------------------------------------------------------------
__global__ void gcn_zero_f32(float* __restrict__ p, int n) {
    int i = blockIdx.x * blockDim.x + threadIdx.x;
    if (i < n) p[i] = 0.0f;
}

// ---------------------------------------------------------------------------
// Kernel 1: hp[N,64] = h[N,256] @ W[256,64] via V_WMMA_F32_16X16X4_F32
//   block = 128 threads = 4 waves; each wave owns a 16-row stripe and
//   computes ALL four 16-wide N tiles (4 accumulators), so each A fragment
//   feeds 4 WMMAs and h is read exactly once per block. Block covers 64 rows.
//   W (64 KB) staged into LDS via gfx1250 GLOBAL_LOAD_ASYNC_TO_LDS_B128
//   (ASYNCcnt) -- WGP has 320 KB LDS.
//
// FP32 WMMA fragment layouts (ISA 7.12.2, wave32):
//   A (16x4):  lane L<16  -> M=L,    VGPR0=K0, VGPR1=K1
//              lane L>=16 -> M=L-16, VGPR0=K2, VGPR1=K3
//   B (4x16):  VGPR v, lane<16 -> K=v,  N=lane ; lane>=16 -> K=v+2, N=lane-16
//   C/D(16x16):VGPR v, lane<16 -> M=v,  N=lane ; lane>=16 -> M=v+8, N=lane-16
// ---------------------------------------------------------------------------
__global__ __launch_bounds__(128)
void gcn_gemm_wmma(const float* __restrict__ h,
                   const float* __restrict__ W,
                   float* __restrict__ hp,
                   int n_rows)
{
    __shared__ float sW[IN_FEA * OU_FEA]; // 64 KB

    const int tid = threadIdx.x;

    // ---- async stage of W into LDS (CDNA5 async-to-LDS path, ASYNCcnt) ----
    {
        const uint32_t lds0  = (uint32_t)(uintptr_t)(&sW[0]); // low 32b = LDS off
        const char*    gbase = (const char*)W;
        #pragma unroll 4
        for (int c = tid; c < (IN_FEA * OU_FEA * 4) / 16; c += 128) {
            uint32_t lds_off = lds0 + (uint32_t)c * 16u;
            uint64_t gaddr   = (uint64_t)(uintptr_t)(gbase + (size_t)c * 16);
            asm volatile("global_load_async_to_lds_b128 %0, %1, off"
                         :: "v"(lds_off), "v"(gaddr) : "memory");
        }
        asm volatile("s_wait_asynccnt 0" ::: "memory");
    }
    __syncthreads();

    const int wave  = tid >> 5;        // 0..3 -> row stripe
    const int lane  = tid & 31;
    const int l16   = lane & 15;
    const int hi    = lane >> 4;       // 0 or 1
    const int khalf = hi * 2;          // K sub-offset 0 or 2

    const int m0 = blockIdx.x * 64 + wave * 16;

    // clamp load row so EXEC stays all-ones for WMMA on the ragged edge
    int mrow = m0 + l16;
    if (mrow >= n_rows) mrow = n_rows - 1;

    const float* __restrict__ hrow  = h + (size_t)mrow * IN_FEA + khalf;
    const float* __restrict__ bbase = sW + l16 + khalf * OU_FEA;

    v8f acc0 = {}, acc1 = {}, acc2 = {}, acc3 = {};
    #pragma unroll 4
    for (int k = 0; k < IN_FEA; k += 4) {
        v2f a;                                  // contiguous -> one b64 load
        a.x = hrow[k + 0];
        a.y = hrow[k + 1];
        const float* bk = bbase + k * OU_FEA;
        v2f b0; b0.x = bk[ 0]; b0.y = bk[ 0 + OU_FEA];
        v2f b1; b1.x = bk[16]; b1.y = bk[16 + OU_FEA];
        v2f b2; b2.x = bk[32]; b2.y = bk[32 + OU_FEA];
        v2f b3; b3.x = bk[48]; b3.y = bk[48 + OU_FEA];
        acc0 = __builtin_amdgcn_wmma_f32_16x16x4_f32(false, a, false, b0, (short)0, acc0, false, false);
        acc1 = __builtin_amdgcn_wmma_f32_16x16x4_f32(false, a, false, b1, (short)0, acc1, false, false);
        acc2 = __builtin_amdgcn_wmma_f32_16x16x4_f32(false, a, false, b2, (short)0, acc2, false, false);
        acc3 = __builtin_amdgcn_wmma_f32_16x16x4_f32(false, a, false, b3, (short)0, acc3, false, false);
    }

    const int mbase = m0 + hi * 8;
    if (m0 + 16 <= n_rows) {
        // fast path: full tile, unguarded coalesced stores
        #pragma unroll
        for (int v = 0; v < 8; ++v) {
            float* drow = hp + (size_t)(mbase + v) * OU_FEA + l16;
            drow[ 0] = acc0[v];
            drow[16] = acc1[v];
            drow[32] = acc2[v];
            drow[48] = acc3[v];
        }
    } else {
        #pragma unroll
        for (int v = 0; v < 8; ++v) {
            int m = mbase + v;
            if (m < n_rows) {
                float* drow = hp + (size_t)m * OU_FEA + l16;
                drow[ 0] = acc0[v];
                drow[16] = acc1[v];
                drow[32] = acc2[v];
                drow[48] = acc3[v];
            }
        }
    }
}

// ---------------------------------------------------------------------------
// Kernel 2: out[col[e]] += vals[e] * hp[row[e]]   (per-edge atomic scatter)
//   one wave per edge; lane owns columns {2*lane, 2*lane+1} -> coalesced
//   float2 gather of the 256B hp row, then 2x global_atomic_add_f32.
//   hp and out are L2-resident (12.8 MB each vs 192 MB L2), so the random
//   traffic stays on-chip and atomics resolve at L2.
// ---------------------------------------------------------------------------
__global__ __launch_bounds__(256)
void gcn_scatter(const float* __restrict__ hp,
                 const int* __restrict__ row,
                 const int* __restrict__ col,
                 const float* __restrict__ vals,
                 float* __restrict__ out,
                 int E)
{
    const int e = blockIdx.x * (blockDim.x >> 5) + (threadIdx.x >> 5);
    if (e >= E) return;
    const int lane = threadIdx.x & 31;

    const int   r = row[e];
    const int   c = col[e];
    const float v = vals[e];

    const float2 m = ((const float2*)(hp + (size_t)r * OU_FEA))[lane];
    float* dst = out + (size_t)c * OU_FEA + lane * 2;
    atomicAdd(dst + 0, v * m.x);
    atomicAdd(dst + 1, v * m.y);
}

// ---------------------------------------------------------------------------
extern "C" void kernel_launch(void* const* d_in, const int* in_sizes, int n_in,
                              void* d_out, int out_size, void* d_ws, size_t ws_size,
                              hipStream_t stream) {
    const float* h    = (const float*)d_in[0];   // [N, 256]
    const float* W    = (const float*)d_in[1];   // [256, 64]
    const int*   row  = (const int*)d_in[2];     // [E]
    const int*   col  = (const int*)d_in[3];     // [E]
    const float* vals = (const float*)d_in[4];   // [E]
    float*       out  = (float*)d_out;           // [N, 64]
    float*       hp   = (float*)d_ws;            // scratch [N, 64] = 12.8 MB

    const int N = in_sizes[0] / IN_FEA;
    const int E = in_sizes[2];

    gcn_zero_f32<<<(out_size + 255) / 256, 256, 0, stream>>>(out, out_size);

    gcn_gemm_wmma<<<(N + 63) / 64, 128, 0, stream>>>(h, W, hp, N);

    const int edgesPerBlock = 256 / 32; // 8 waves per block, 1 edge per wave
    gcn_scatter<<<(E + edgesPerBlock - 1) / edgesPerBlock, 256, 0, stream>>>(
        hp, row, col, vals, out, E);
}